// GCN_36223754174562
// MI455X (gfx1250) — compile-verified
//
#include <hip/hip_runtime.h>

#define D 128
#define NGRAPHS 512
#define ASTRIDE 130  // LDS row stride in floats (pads away bank conflicts)

typedef __attribute__((ext_vector_type(2))) float v2f;
typedef __attribute__((ext_vector_type(8))) float v8f;

// ---------------- degree / normalization ----------------
__global__ void k_init_deg(float* __restrict__ deg, int N) {
    int i = blockIdx.x * blockDim.x + threadIdx.x;
    if (i < N) deg[i] = 1.0f;  // self loop contributes 1
}

__global__ void k_count_deg(const int* __restrict__ col, float* __restrict__ deg, int E) {
    int i = blockIdx.x * blockDim.x + threadIdx.x;
    if (i < E) atomicAdd(&deg[col[i]], 1.0f);
}

__global__ void k_dinv(const float* __restrict__ deg, float* __restrict__ dinv, int N) {
    int i = blockIdx.x * blockDim.x + threadIdx.x;
    if (i < N) dinv[i] = rsqrtf(fmaxf(deg[i], 1.0f));
}

// ---------------- W transpose (so B fragment is a contiguous float2) ----------------
__global__ void k_transpose_w(const float* __restrict__ W, float* __restrict__ Wt) {
    int i = blockIdx.x * blockDim.x + threadIdx.x;  // 0 .. D*D-1
    int k = i >> 7, c = i & (D - 1);
    Wt[c * D + k] = W[i];
}

// ---------------- GEMM: out[N x D] = (relu?)A[N x D] @ W  via V_WMMA_F32_16X16X4_F32 ----------------
__global__ __launch_bounds__(256) void k_gemm(const float* __restrict__ A,
                                              const float* __restrict__ Wt,  // [D x D] transposed
                                              float* __restrict__ out, int N, int relu_in) {
    __shared__ __align__(16) float lA[16 * ASTRIDE];
    const int t = threadIdx.x;
    const int rowbase = blockIdx.x << 4;

    // Stage 16xD A-tile into LDS (coalesced global read), fuse ReLU of previous layer.
    for (int i = t; i < 16 * D; i += 256) {
        int r = i >> 7, k = i & (D - 1);
        int row = rowbase + r;
        float v = (row < N) ? A[(size_t)row * D + k] : 0.0f;
        if (relu_in) v = fmaxf(v, 0.0f);
        lA[r * ASTRIDE + k] = v;
    }
    __syncthreads();

    const int lane  = t & 31;
    const int l16   = lane & 15;
    const int khalf = (lane >> 4) << 1;   // lanes 0-15: K offset 0,1 ; lanes 16-31: K offset 2,3
    const int colbase = (t >> 5) << 4;    // wave id -> 16-column block of W

    const float* pa = &lA[l16 * ASTRIDE + khalf];
    const float* pb = &Wt[(size_t)(colbase + l16) * D + khalf];

    v8f acc = {};
#pragma unroll
    for (int k = 0; k < D; k += 4) {
        v2f a = *(const v2f*)(pa + k);   // ds_load_b64, conflict-free (stride 130)
        v2f b = *(const v2f*)(pb + k);   // global_load_b64, L2-resident W
        acc = __builtin_amdgcn_wmma_f32_16x16x4_f32(false, a, false, b, (short)0, acc,
                                                    false, false);
    }

    const int mhi = (lane >> 4) << 3;    // lanes>=16 hold M = r+8
#pragma unroll
    for (int r = 0; r < 8; ++r) {
        int row = rowbase + r + mhi;
        if (row < N) out[(size_t)row * D + colbase + l16] = acc[r];
    }
}

// ---------------- h_next = hw * dinv^2 (self loop) + bias ----------------
__global__ void k_selfloop_bias(const float* __restrict__ hw, const float* __restrict__ dinv,
                                const float* __restrict__ b, float* __restrict__ hn, int N) {
    size_t i = (size_t)blockIdx.x * blockDim.x + threadIdx.x;
    if (i >= (size_t)N * D) return;
    int n = (int)(i >> 7);
    int d = (int)(i & (D - 1));
    float s = dinv[n];
    hn[i] = hw[i] * s * s + b[d];
}

// ---------------- edge scatter: one wave per edge, float4 per lane, atomics into L2 ----------------
__global__ __launch_bounds__(256) void k_edges(const float* __restrict__ hw,
                                               const int* __restrict__ row,
                                               const int* __restrict__ colv,
                                               const float* __restrict__ dinv,
                                               float* __restrict__ hn, int E) {
    size_t gid = (size_t)blockIdx.x * blockDim.x + threadIdx.x;
    int e = (int)(gid >> 5);
    if (e >= E) return;
    int lane = (int)(gid & 31);
    int r = row[e], c = colv[e];
    float nrm = dinv[r] * dinv[c];
    float4 v = *(const float4*)(hw + (size_t)r * D + lane * 4);
    float* dst = hn + (size_t)c * D + lane * 4;
    atomicAdd(dst + 0, v.x * nrm);
    atomicAdd(dst + 1, v.y * nrm);
    atomicAdd(dst + 2, v.z * nrm);
    atomicAdd(dst + 3, v.w * nrm);
}

// ---------------- pooling ----------------
__global__ void k_pool_zero(float* __restrict__ pooled, float* __restrict__ cnt) {
    int i = blockIdx.x * blockDim.x + threadIdx.x;
    if (i < NGRAPHS * D) pooled[i] = 0.0f;
    if (i < NGRAPHS) cnt[i] = 0.0f;
}

__global__ __launch_bounds__(256) void k_pool_accum(const float* __restrict__ h,
                                                    const int* __restrict__ batch,
                                                    float* __restrict__ pooled,
                                                    float* __restrict__ cnt, int N) {
    size_t gid = (size_t)blockIdx.x * blockDim.x + threadIdx.x;
    int n = (int)(gid >> 5);
    if (n >= N) return;
    int lane = (int)(gid & 31);
    int g = batch[n];
    float4 v = *(const float4*)(h + (size_t)n * D + lane * 4);  // final-layer ReLU fused here
    float* dst = pooled + (size_t)g * D + lane * 4;
    atomicAdd(dst + 0, fmaxf(v.x, 0.0f));
    atomicAdd(dst + 1, fmaxf(v.y, 0.0f));
    atomicAdd(dst + 2, fmaxf(v.z, 0.0f));
    atomicAdd(dst + 3, fmaxf(v.w, 0.0f));
    if (lane == 0) atomicAdd(&cnt[g], 1.0f);
}

// ---------------- final linear: one wave per graph, wave32 shuffle reduction ----------------
__global__ __launch_bounds__(256) void k_final(const float* __restrict__ pooled,
                                               const float* __restrict__ cnt,
                                               const float* __restrict__ Wl,
                                               const float* __restrict__ bl,
                                               float* __restrict__ out) {
    int t = threadIdx.x;
    int lane = t & 31;
    int g = blockIdx.x * (blockDim.x >> 5) + (t >> 5);
    if (g >= NGRAPHS) return;
    float4 p = *(const float4*)(pooled + (size_t)g * D + lane * 4);
    float4 w = *(const float4*)(Wl + lane * 4);
    float dot = p.x * w.x + p.y * w.y + p.z * w.z + p.w * w.w;
#pragma unroll
    for (int off = 16; off > 0; off >>= 1) dot += __shfl_xor(dot, off, 32);
    if (lane == 0) out[g] = dot / fmaxf(cnt[g], 1.0f) + bl[0];
}

// ---------------- driver ----------------
extern "C" void kernel_launch(void* const* d_in, const int* in_sizes, int n_in,
                              void* d_out, int out_size, void* d_ws, size_t ws_size,
                              hipStream_t stream) {
    const float* x     = (const float*)d_in[0];
    const int*   ei    = (const int*)d_in[1];
    const int*   batch = (const int*)d_in[2];
    const float* Ws    = (const float*)d_in[3];
    const float* bs    = (const float*)d_in[4];
    const float* Wl    = (const float*)d_in[5];
    const float* bl    = (const float*)d_in[6];
    float* out = (float*)d_out;

    const int N = in_sizes[2];
    const int E = in_sizes[1] / 2;
    const int* rowp = ei;
    const int* colp = ei + E;

    size_t nd = (size_t)((N + 15) & ~15) * D;
    float* bufA   = (float*)d_ws;        // hw = h @ W
    float* bufB   = bufA + nd;           // h_next (post-aggregation)
    float* deg    = bufB + nd;
    float* dinv   = deg + N;
    float* Wt     = dinv + N;            // transposed layer weight
    float* pooled = Wt + D * D;
    float* cnt    = pooled + NGRAPHS * D;

    dim3 B(256);

    k_init_deg<<<(N + 255) / 256, B, 0, stream>>>(deg, N);
    k_count_deg<<<(E + 255) / 256, B, 0, stream>>>(colp, deg, E);
    k_dinv<<<(N + 255) / 256, B, 0, stream>>>(deg, dinv, N);

    const float* hin = x;
    for (int l = 0; l < 4; ++l) {
        k_transpose_w<<<(D * D) / 256, B, 0, stream>>>(Ws + (size_t)l * D * D, Wt);
        k_gemm<<<(N + 15) / 16, B, 0, stream>>>(hin, Wt, bufA, N, l > 0 ? 1 : 0);
        unsigned nb = (unsigned)(((size_t)N * D + 255) / 256);
        k_selfloop_bias<<<nb, B, 0, stream>>>(bufA, dinv, bs + (size_t)l * D, bufB, N);
        unsigned eb = (unsigned)(((size_t)E * 32 + 255) / 256);
        k_edges<<<eb, B, 0, stream>>>(bufA, rowp, colp, dinv, bufB, E);
        hin = bufB;  // next layer reads bufB (GEMM applies ReLU on load)
    }

    k_pool_zero<<<(NGRAPHS * D + 255) / 256, B, 0, stream>>>(pooled, cnt);
    unsigned pb = (unsigned)(((size_t)N * 32 + 255) / 256);
    k_pool_accum<<<pb, B, 0, stream>>>(bufB, batch, pooled, cnt, N);
    k_final<<<(NGRAPHS + 7) / 8, B, 0, stream>>>(pooled, cnt, Wl, bl, out);
}